// Attention_85925115723783
// MI455X (gfx1250) — compile-verified
//
#include <hip/hip_runtime.h>

// Paged GQA decode attention, MI455X (gfx1250, wave32).
// Memory-bound (~2 FLOP/byte): split-K flash decode, WMMA f32 16x16x4 for QK,
// coalesced b128 streaming of K (via padded LDS) and V (direct), VALU PV.

#define B_     32
#define H_     32
#define HKV_   8
#define G_     4
#define D_     128
#define P_     16
#define M_     128
#define LMAX_  2048
#define SPLITS 8
#define CHUNK  (LMAX_ / SPLITS)   // 256 tokens = 16 pages per split
#define PAIRS  (B_ * HKV_)        // 256
#define SCALE_ 0.08838834764831845f
#define LDS_STRIDE 132            // dwords per LDS row (128 + 4 pad) -> conflict-free b64 reads

typedef float v2f __attribute__((ext_vector_type(2)));
typedef float v8f __attribute__((ext_vector_type(8)));

__global__ __launch_bounds__(32) void pa_split_kernel(
    const float* __restrict__ q,        // [B, H, D]
    const float* __restrict__ kin,      // [B, HKV, D]
    const float* __restrict__ vin,      // [B, HKV, D]
    const float* __restrict__ k_cache,  // [NUM_PAGES, P, D]
    const float* __restrict__ v_cache,  // [NUM_PAGES, P, D]
    const int*   __restrict__ bh_seq_lens,   // [SLOTS, HKV]
    const int*   __restrict__ page_table,    // [SLOTS, HKV, M]
    const int*   __restrict__ batch_mapping, // [B]
    float* __restrict__ PO,   // [PAIRS, SPLITS, G, D]
    float* __restrict__ PM,   // [PAIRS, SPLITS, G]
    float* __restrict__ PL)   // [PAIRS, SPLITS, G]
{
  __shared__ __align__(16) float lds[2 * 16 * LDS_STRIDE];
  float* ldsK = lds;
  float* ldsQ = lds + 16 * LDS_STRIDE;

  const int bid  = blockIdx.x;
  const int pair = bid >> 3;            // / SPLITS
  const int s    = bid & (SPLITS - 1);
  const int b    = pair >> 3;           // / HKV
  const int h    = pair & (HKV_ - 1);
  const int lane = threadIdx.x;

  const int slot = batch_mapping[b];
  const int Lpos = bh_seq_lens[slot * HKV_ + h];     // new token goes at position Lpos
  const int* pt  = page_table + (slot * HKV_ + h) * M_;

  float* po = PO + (size_t)(pair * SPLITS + s) * (G_ * D_);
  float* pm = PM + (size_t)(pair * SPLITS + s) * G_;
  float* pl = PL + (size_t)(pair * SPLITS + s) * G_;

  const int t0 = s * CHUNK;
  if (t0 > Lpos) {
    // empty split: weight will be exp(-1e30 - M) == 0 in combine
    #pragma unroll
    for (int g = 0; g < G_; ++g)
      ((float4*)(po + g * D_))[lane] = make_float4(0.f, 0.f, 0.f, 0.f);
    if (lane == 0) {
      #pragma unroll
      for (int g = 0; g < G_; ++g) { pm[g] = -1e30f; pl[g] = 0.f; }
    }
    return;
  }
  const int t1 = min(t0 + CHUNK, Lpos + 1);

  // ---- stage pre-scaled Q into LDS: rows 0..3 = groups, rows 4..15 = zero ----
  {
    const float4* q4 = (const float4*)(q + (size_t)(b * H_ + h * G_) * D_);
    #pragma unroll
    for (int i = 0; i < 16; ++i) {
      int idx = i * 32 + lane;          // 512 float4 total
      int row = idx >> 5, c4 = idx & 31;
      float4 val = make_float4(0.f, 0.f, 0.f, 0.f);
      if (row < G_) {
        float4 t = q4[row * 32 + c4];
        val = make_float4(t.x * SCALE_, t.y * SCALE_, t.z * SCALE_, t.w * SCALE_);
      }
      ((float4*)ldsQ)[row * 33 + c4] = val;   // 33 float4 = 132-dword padded row
    }
  }

  const int hh = lane >> 4;     // half: 0 -> K-chunk dims {0,1}/tokens r..r ; 1 -> dims {2,3}
  const int tq = lane & 15;     // A row (token) == B column (group) index

  float  m_run[G_], l_run[G_];
  float4 oac[G_];               // lane owns output dims [4*lane, 4*lane+3] for each group
  #pragma unroll
  for (int g = 0; g < G_; ++g) {
    m_run[g] = -1e30f; l_run[g] = 0.f;
    oac[g] = make_float4(0.f, 0.f, 0.f, 0.f);
  }

  const float4* kin4 = (const float4*)(kin + (size_t)(b * HKV_ + h) * D_);
  const float4* vin4 = (const float4*)(vin + (size_t)(b * HKV_ + h) * D_);

  for (int base = t0; base < t1; base += P_) {
    const int page_id = pt[base >> 4];
    const float4* gk = (const float4*)(k_cache + (size_t)page_id * (P_ * D_));
    const float4* gv = (const float4*)(v_cache + (size_t)page_id * (P_ * D_));

    __syncthreads();  // WAR vs previous tile's fragment reads
    // ---- stage 8KB K tile into padded LDS (coalesced b128 loads) ----
    #pragma unroll
    for (int i = 0; i < 16; ++i) {
      int idx = i * 32 + lane;
      int row = idx >> 5, c4 = idx & 31;
      ((float4*)ldsK)[row * 33 + c4] = gk[idx];
    }
    // substitute the new token's K row (reference scatters k into the cache at Lpos)
    if (Lpos >= base && Lpos < base + P_) {
      ((float4*)ldsK)[(Lpos - base) * 33 + lane] = kin4[lane];
    }
    __syncthreads();

    // ---- S^T(16 tokens x 16 cols) = K_tile(16x128) * Qs(128x16) via 32 chained WMMAs ----
    v8f acc = {};
    #pragma unroll
    for (int j = 0; j < 32; ++j) {
      int off = 4 * j + 2 * hh;                                // head-dim chunk, per-half
      v2f a  = *(const v2f*)(ldsK + tq * LDS_STRIDE + off);    // A: token row tq
      v2f bq = *(const v2f*)(ldsQ + tq * LDS_STRIDE + off);    // B: group column tq
      acc = __builtin_amdgcn_wmma_f32_16x16x4_f32(
          /*neg_a=*/false, a, /*neg_b=*/false, bq,
          /*c_mod=*/(short)0, acc, /*reuse_a=*/false, /*reuse_b=*/false);
    }

    // ---- per-lane scores: token = base + 8*hh + r, column g = tq ----
    float sv[8];
    float lmax = -1e30f;
    #pragma unroll
    for (int r = 0; r < 8; ++r) {
      int t = base + 8 * hh + r;
      sv[r] = (t <= Lpos) ? acc[r] : -1e30f;
      lmax = fmaxf(lmax, sv[r]);
    }
    lmax = fmaxf(lmax, __shfl_xor(lmax, 16, 32));   // join the two token halves
    float mnew[G_], cfac[G_];
    #pragma unroll
    for (int g = 0; g < G_; ++g) {
      float tm = __shfl(lmax, g, 32);               // column-g max lives in lane g
      mnew[g] = fmaxf(m_run[g], tm);
      cfac[g] = __expf(m_run[g] - mnew[g]);
    }

    float pv[8];
    float psum = 0.f;
    float mloc = mnew[tq & 3];                      // lanes with tq>=4 compute ignored garbage
    #pragma unroll
    for (int r = 0; r < 8; ++r) { pv[r] = __expf(sv[r] - mloc); psum += pv[r]; }
    psum += __shfl_xor(psum, 16, 32);

    #pragma unroll
    for (int g = 0; g < G_; ++g) {
      float ts = __shfl(psum, g, 32);
      l_run[g] = l_run[g] * cfac[g] + ts;
      m_run[g] = mnew[g];
      oac[g].x *= cfac[g]; oac[g].y *= cfac[g];
      oac[g].z *= cfac[g]; oac[g].w *= cfac[g];
    }

    // ---- PV: stream V rows (coalesced b128), broadcast p, FMA into dim-layout acc ----
    #pragma unroll
    for (int tok = 0; tok < P_; ++tok) {
      int t = base + tok;
      if (t < t1) {
        int src = (tok >= 8) ? 16 : 0;
        float p0 = __shfl(pv[tok & 7], src + 0, 32);
        float p1 = __shfl(pv[tok & 7], src + 1, 32);
        float p2 = __shfl(pv[tok & 7], src + 2, 32);
        float p3 = __shfl(pv[tok & 7], src + 3, 32);
        float4 vv = (t == Lpos) ? vin4[lane] : gv[tok * 32 + lane];
        oac[0].x += p0 * vv.x; oac[0].y += p0 * vv.y; oac[0].z += p0 * vv.z; oac[0].w += p0 * vv.w;
        oac[1].x += p1 * vv.x; oac[1].y += p1 * vv.y; oac[1].z += p1 * vv.z; oac[1].w += p1 * vv.w;
        oac[2].x += p2 * vv.x; oac[2].y += p2 * vv.y; oac[2].z += p2 * vv.z; oac[2].w += p2 * vv.w;
        oac[3].x += p3 * vv.x; oac[3].y += p3 * vv.y; oac[3].z += p3 * vv.z; oac[3].w += p3 * vv.w;
      }
    }
  }

  // ---- write partials (unnormalized o, running m, running l) ----
  #pragma unroll
  for (int g = 0; g < G_; ++g)
    ((float4*)(po + g * D_))[lane] = oac[g];
  if (lane == 0) {
    #pragma unroll
    for (int g = 0; g < G_; ++g) { pm[g] = m_run[g]; pl[g] = l_run[g]; }
  }
}

__global__ __launch_bounds__(128) void pa_combine_kernel(
    const float* __restrict__ PO, const float* __restrict__ PM,
    const float* __restrict__ PL, float* __restrict__ out)
{
  const int pair = blockIdx.x;
  const int d    = threadIdx.x;
  const int b    = pair >> 3;
  const int h    = pair & (HKV_ - 1);

  #pragma unroll
  for (int g = 0; g < G_; ++g) {
    float M = -1e30f;
    #pragma unroll
    for (int s = 0; s < SPLITS; ++s)
      M = fmaxf(M, PM[(pair * SPLITS + s) * G_ + g]);
    float Lsum = 0.f, osum = 0.f;
    #pragma unroll
    for (int s = 0; s < SPLITS; ++s) {
      int idx = (pair * SPLITS + s) * G_ + g;
      float w = __expf(PM[idx] - M);     // empty splits: exp(-1e30 - M) == 0
      Lsum += PL[idx] * w;
      osum += PO[(size_t)idx * D_ + d] * w;
    }
    out[(size_t)(b * H_ + h * G_ + g) * D_ + d] = osum / Lsum;
  }
}

extern "C" void kernel_launch(void* const* d_in, const int* in_sizes, int n_in,
                              void* d_out, int out_size, void* d_ws, size_t ws_size,
                              hipStream_t stream) {
  const float* q  = (const float*)d_in[0];
  const float* k  = (const float*)d_in[1];
  const float* v  = (const float*)d_in[2];
  const float* kc = (const float*)d_in[3];
  const float* vc = (const float*)d_in[4];
  const int* lens = (const int*)d_in[5];
  const int* pt   = (const int*)d_in[6];
  const int* bm   = (const int*)d_in[7];
  float* out = (float*)d_out;

  float* PO = (float*)d_ws;                                   // 256*8*4*128 f32 = 4 MiB
  float* PM = PO + (size_t)PAIRS * SPLITS * G_ * D_;          // 8192 f32
  float* PL = PM + (size_t)PAIRS * SPLITS * G_;               // 8192 f32

  pa_split_kernel<<<PAIRS * SPLITS, 32, 0, stream>>>(
      q, k, v, kc, vc, lens, pt, bm, PO, PM, PL);
  pa_combine_kernel<<<PAIRS, 128, 0, stream>>>(PO, PM, PL, out);
}